// GraphTransformer_19636590477895
// MI455X (gfx1250) — compile-verified
//
#include <hip/hip_runtime.h>
#include <hip/hip_bf16.h>
#include <math.h>

// ---------------- problem constants (match reference) ----------------
#define NN    20000
#define EE    320000
#define GG    16
#define D_IN  32
#define D_LAP 16
#define DD    256
#define HH    8
#define DHD   32
#define LL    2
#define DFF   512

typedef __attribute__((ext_vector_type(2))) float v2f;
typedef __attribute__((ext_vector_type(8))) float v8f;

// =====================================================================
// fp32 WMMA GEMM, compile-time K / Ncols so the fully-unrolled K loop
// becomes pure immediate-offset global loads feeding v_wmma_f32_16x16x4_f32.
//   C[M,NCOLS] = A[M,KDIM] * B[KDIM,NCOLS] (+bias)(+resid)(relu)
// One wave computes a 32x32 tile = 2x2 accumulators of 16x16.
// A frag (16x4 f32): lanes 0-15 -> M=lane, VGPR0=K0,VGPR1=K1; lanes 16-31 -> K2,K3.
// B frag (4x16 f32): mirrored with lane = N column.
// C/D (16x16 f32): VGPR r -> M=r (lanes 0-15) / M=r+8 (lanes 16-31), N=lane&15.
// =====================================================================
template <int KDIM, int NCOLS, bool BIAS, bool RESID, bool RELU>
__global__ __launch_bounds__(256) void gemm_wmma_f32(
    const float* __restrict__ A, const float* __restrict__ B,
    const float* __restrict__ bias, const float* __restrict__ resid,
    float* __restrict__ C, int M)
{
    const int lane  = threadIdx.x & 31;
    const int wid   = blockIdx.x * (blockDim.x >> 5) + (threadIdx.x >> 5);
    constexpr int ntile = NCOLS >> 5;
    const int tileM = (wid / ntile) * 32;
    const int tileN = (wid % ntile) * 32;
    if (tileM >= M) return;                 // wave-uniform; EXEC stays all-ones

    const int half = lane >> 4;             // selects K pair (0,1) vs (2,3)
    const int l15  = lane & 15;

    const float* A0 = A + (size_t)(tileM + l15) * KDIM + 2 * half;
    const float* A1 = A0 + (size_t)16 * KDIM;
    const float* B0 = B + (size_t)(2 * half) * NCOLS + (tileN + l15);

    v8f c00 = {}; v8f c01 = {}; v8f c10 = {}; v8f c11 = {};

    #pragma unroll
    for (int k = 0; k < KDIM; k += 4) {
        v2f a0 = *(const v2f*)(A0 + k);
        v2f a1 = *(const v2f*)(A1 + k);
        const float* bp = B0 + k * NCOLS;   // compile-time multiple -> imm offsets
        v2f b0, b1;
        b0.x = bp[0];         b0.y = bp[NCOLS];
        b1.x = bp[16];        b1.y = bp[NCOLS + 16];
        c00 = __builtin_amdgcn_wmma_f32_16x16x4_f32(false, a0, false, b0, (short)0, c00, false, false);
        c01 = __builtin_amdgcn_wmma_f32_16x16x4_f32(false, a0, false, b1, (short)0, c01, false, false);
        c10 = __builtin_amdgcn_wmma_f32_16x16x4_f32(false, a1, false, b0, (short)0, c10, false, false);
        c11 = __builtin_amdgcn_wmma_f32_16x16x4_f32(false, a1, false, b1, (short)0, c11, false, false);
    }

    const int rowBase = tileM + 8 * half;
    auto storeTile = [&](v8f c, int mOff, int nOff) {
        const int col = tileN + nOff + l15;
        const float bv = BIAS ? bias[col] : 0.0f;
        #pragma unroll
        for (int r = 0; r < 8; ++r) {
            const int row = rowBase + mOff + r;
            float v = c[r] + bv;
            if (RESID) v += resid[(size_t)row * NCOLS + col];
            if (RELU)  v = v > 0.0f ? v : 0.0f;
            C[(size_t)row * NCOLS + col] = v;
        }
    };
    storeTile(c00, 0, 0);  storeTile(c01, 0, 16);
    storeTile(c10, 16, 0); storeTile(c11, 16, 16);
}

// ---------------- input embedding: h = feat@Wh + bh + (lap*sf)@Wlp + blp ----
__global__ __launch_bounds__(256) void embed_kernel(
    const float* __restrict__ feat, const float* __restrict__ lap,
    const float* __restrict__ sf,   const float* __restrict__ Wh,
    const float* __restrict__ bh,   const float* __restrict__ Wlp,
    const float* __restrict__ blp,  float* __restrict__ h)
{
    const int idx = blockIdx.x * blockDim.x + threadIdx.x;  // n*DD + d
    const int n = idx >> 8, d = idx & 255;
    float acc = bh[d] + blp[d];
    const float* f = feat + (size_t)n * D_IN;
    #pragma unroll
    for (int k = 0; k < D_IN; ++k) acc += f[k] * Wh[k * DD + d];
    const float* lp = lap + (size_t)n * D_LAP;
    #pragma unroll
    for (int k = 0; k < D_LAP; ++k) acc += lp[k] * sf[k] * Wlp[k * DD + d];
    h[idx] = acc;
}

// ------------- per-edge score: s = exp(clip(K[src].Q[dst]/sqrt(dh))); seg max
__global__ __launch_bounds__(256) void edge_score_kernel(
    const float* __restrict__ Q, const float* __restrict__ Kmat,
    const int* __restrict__ src, const int* __restrict__ dst,
    float* __restrict__ s, unsigned int* __restrict__ mbits)
{
    const int idx = blockIdx.x * blockDim.x + threadIdx.x;  // e*HH + h
    const int e = idx >> 3, hh = idx & 7;
    const int sN = src[e], dN = dst[e];
    const float* kp = Kmat + (size_t)sN * DD + hh * DHD;
    const float* qp = Q    + (size_t)dN * DD + hh * DHD;
    float acc = 0.0f;
    #pragma unroll
    for (int i = 0; i < DHD; ++i) acc += kp[i] * qp[i];
    float sc = acc * 0.1767766952966369f;               // 1/sqrt(32)
    sc = fminf(fmaxf(sc, -5.0f), 5.0f);
    const float sv = expf(sc);                          // sv > 0 always
    s[idx] = sv;
    // positive floats are order-isomorphic to their uint bit patterns
    atomicMax(&mbits[dN * HH + hh], __float_as_uint(sv));
}

// ------------- p = exp(s - m[dst]); den += p; msg += p * V[src] -------------
__global__ __launch_bounds__(256) void edge_aggr_kernel(
    const float* __restrict__ V, const int* __restrict__ src,
    const int* __restrict__ dst, const float* __restrict__ s,
    const float* __restrict__ m, float* __restrict__ den, float* __restrict__ msg)
{
    const int idx = blockIdx.x * blockDim.x + threadIdx.x;
    const int e = idx >> 3, hh = idx & 7;
    const int dN = dst[e];
    const float p = expf(s[idx] - m[dN * HH + hh]);
    atomicAdd(&den[dN * HH + hh], p);
    const float* vp = V + (size_t)src[e] * DD + hh * DHD;
    float* mp = msg + (size_t)dN * DD + hh * DHD;
    #pragma unroll
    for (int i = 0; i < DHD; ++i) atomicAdd(&mp[i], p * vp[i]);
}

// ------------- attn = msg / max(den, 1-if-zero), in place -------------------
__global__ __launch_bounds__(256) void attn_div_kernel(
    float* __restrict__ msg, const float* __restrict__ den)
{
    const int idx = blockIdx.x * blockDim.x + threadIdx.x;  // n*DD + d
    const int n = idx >> 8, hh = (idx & 255) >> 5;
    const float dv = den[n * HH + hh];
    msg[idx] = msg[idx] / (dv > 0.0f ? dv : 1.0f);
}

// ------------- LayerNorm: one wave32 per row of 256 -------------------------
__global__ __launch_bounds__(256) void layernorm_kernel(
    const float* __restrict__ x, const float* __restrict__ g,
    const float* __restrict__ b, float* __restrict__ out)
{
    const int row  = blockIdx.x * (blockDim.x >> 5) + (threadIdx.x >> 5);
    const int lane = threadIdx.x & 31;
    const float* xr = x + (size_t)row * DD;
    float v[8], sum = 0.0f, sq = 0.0f;
    #pragma unroll
    for (int i = 0; i < 8; ++i) { v[i] = xr[lane + 32 * i]; sum += v[i]; sq += v[i] * v[i]; }
    #pragma unroll
    for (int off = 16; off; off >>= 1) {
        sum += __shfl_xor(sum, off, 32);
        sq  += __shfl_xor(sq,  off, 32);
    }
    const float mean = sum * (1.0f / DD);
    const float var  = sq * (1.0f / DD) - mean * mean;
    const float rstd = rsqrtf(var + 1e-5f);
    float* o = out + (size_t)row * DD;
    #pragma unroll
    for (int i = 0; i < 8; ++i) {
        const int d = lane + 32 * i;
        o[d] = g[d] * (v[i] - mean) * rstd + b[d];
    }
}

// ------------- readout: per-graph mean ------------------------------------
__global__ __launch_bounds__(256) void readout_acc_kernel(
    const float* __restrict__ h, const int* __restrict__ gid,
    float* __restrict__ sums, int* __restrict__ counts)
{
    const int idx = blockIdx.x * blockDim.x + threadIdx.x;  // n*DD + d
    const int n = idx >> 8, d = idx & 255;
    const int gg = gid[n];
    atomicAdd(&sums[gg * DD + d], h[idx]);
    if (d == 0) atomicAdd(&counts[gg], 1);
}

__global__ __launch_bounds__(256) void readout_div_kernel(
    const float* __restrict__ sums, const int* __restrict__ counts,
    float* __restrict__ out)
{
    const int idx = blockIdx.x * blockDim.x + threadIdx.x;  // g*DD + d
    const int c = counts[idx >> 8];
    out[idx] = sums[idx] / (float)(c > 0 ? c : 1);
}

// =====================================================================
template <int KDIM, int NCOLS, bool BIAS, bool RESID, bool RELU>
static inline void launch_gemm(const float* A, const float* B, const float* bias,
                               const float* resid, float* C, int M, hipStream_t st)
{
    const int waves  = (M / 32) * (NCOLS / 32);
    const int blocks = waves / 8;            // 8 waves (256 thr) per block; exact here
    gemm_wmma_f32<KDIM, NCOLS, BIAS, RESID, RELU>
        <<<blocks, 256, 0, st>>>(A, B, bias, resid, C, M);
}

extern "C" void kernel_launch(void* const* d_in, const int* in_sizes, int n_in,
                              void* d_out, int out_size, void* d_ws, size_t ws_size,
                              hipStream_t stream)
{
    (void)in_sizes; (void)n_in; (void)out_size; (void)ws_size;

    const float* feat = (const float*)d_in[0];
    const float* lap  = (const float*)d_in[1];
    const float* sf   = (const float*)d_in[2];
    const int*   src  = (const int*)d_in[3];
    const int*   dst  = (const int*)d_in[4];
    const int*   gid  = (const int*)d_in[5];
    const float* Wh   = (const float*)d_in[6];
    const float* bh   = (const float*)d_in[7];
    const float* Wlp  = (const float*)d_in[8];
    const float* blp  = (const float*)d_in[9];
    const float* Wq   = (const float*)d_in[10];
    const float* Wk   = (const float*)d_in[11];
    const float* Wv   = (const float*)d_in[12];
    const float* Wo   = (const float*)d_in[13];
    const float* bo   = (const float*)d_in[14];
    const float* g1   = (const float*)d_in[15];
    const float* b1   = (const float*)d_in[16];
    const float* W1   = (const float*)d_in[17];
    const float* c1   = (const float*)d_in[18];
    const float* W2   = (const float*)d_in[19];
    const float* c2   = (const float*)d_in[20];
    const float* g2   = (const float*)d_in[21];
    const float* b2   = (const float*)d_in[22];

    // ---- workspace carve-up (floats); total ~44M floats (~176 MB) ----
    float* W = (float*)d_ws;
    const size_t ND = (size_t)NN * DD;        // 5,120,000
    float* h    = W;
    float* tmp  = W + ND;
    float* Qb   = W + 2 * ND;
    float* Kb   = W + 3 * ND;
    float* Vb   = W + 4 * ND;
    float* msg  = W + 5 * ND;
    float* ff   = W + 6 * ND;                 // N x DFF = 2*ND
    float* s    = W + 8 * ND;                 // E*H
    float* mmax = s + (size_t)EE * HH;        // N*H
    float* den  = mmax + (size_t)NN * HH;     // N*H
    float* sums = den + (size_t)NN * HH;      // G*D
    int*   cnts = (int*)(sums + GG * DD);     // G

    // ---- input embedding ----
    embed_kernel<<<(NN * DD) / 256, 256, 0, stream>>>(feat, lap, sf, Wh, bh, Wlp, blp, h);

    for (int l = 0; l < LL; ++l) {
        const float* Wq_l = Wq + (size_t)l * DD * DD;
        const float* Wk_l = Wk + (size_t)l * DD * DD;
        const float* Wv_l = Wv + (size_t)l * DD * DD;
        const float* Wo_l = Wo + (size_t)l * DD * DD;
        const float* bo_l = bo + (size_t)l * DD;
        const float* g1_l = g1 + (size_t)l * DD;
        const float* b1_l = b1 + (size_t)l * DD;
        const float* W1_l = W1 + (size_t)l * DD * DFF;
        const float* c1_l = c1 + (size_t)l * DFF;
        const float* W2_l = W2 + (size_t)l * DFF * DD;
        const float* c2_l = c2 + (size_t)l * DD;
        const float* g2_l = g2 + (size_t)l * DD;
        const float* b2_l = b2 + (size_t)l * DD;

        // Q/K/V projections (WMMA fp32, K=256 -> N=256)
        launch_gemm<DD, DD, false, false, false>(h, Wq_l, nullptr, nullptr, Qb, NN, stream);
        launch_gemm<DD, DD, false, false, false>(h, Wk_l, nullptr, nullptr, Kb, NN, stream);
        launch_gemm<DD, DD, false, false, false>(h, Wv_l, nullptr, nullptr, Vb, NN, stream);

        // zero segment buffers (graph-capture-safe memset nodes)
        hipMemsetAsync(mmax, 0, (size_t)NN * HH * sizeof(float), stream);
        hipMemsetAsync(den,  0, (size_t)NN * HH * sizeof(float), stream);
        hipMemsetAsync(msg,  0, ND * sizeof(float), stream);

        // edge attention
        edge_score_kernel<<<(EE * HH) / 256, 256, 0, stream>>>(Qb, Kb, src, dst, s,
                                                               (unsigned int*)mmax);
        edge_aggr_kernel<<<(EE * HH) / 256, 256, 0, stream>>>(Vb, src, dst, s, mmax, den, msg);
        attn_div_kernel<<<(NN * DD) / 256, 256, 0, stream>>>(msg, den);

        // output projection + bias + residual, then LN -> h
        launch_gemm<DD, DD, true, true, false>(msg, Wo_l, bo_l, h, tmp, NN, stream);
        layernorm_kernel<<<NN / 8, 256, 0, stream>>>(tmp, g1_l, b1_l, h);

        // FFN: relu(h@W1+c1) @ W2 + c2 + h, then LN -> h
        launch_gemm<DD, DFF, true, false, true>(h, W1_l, c1_l, nullptr, ff, NN, stream);
        launch_gemm<DFF, DD, true, true, false>(ff, W2_l, c2_l, h, tmp, NN, stream);
        layernorm_kernel<<<NN / 8, 256, 0, stream>>>(tmp, g2_l, b2_l, h);
    }

    // ---- mean readout per graph ----
    hipMemsetAsync(sums, 0, (size_t)GG * DD * sizeof(float) + GG * sizeof(int), stream);
    readout_acc_kernel<<<(NN * DD) / 256, 256, 0, stream>>>(h, gid, sums, cnts);
    readout_div_kernel<<<(GG * DD) / 256, 256, 0, stream>>>(sums, cnts, (float*)d_out);
}